// ArcShapeRadiusConfigVisibleNeighDist_37271726195502
// MI455X (gfx1250) — compile-verified
//
#include <hip/hip_runtime.h>

// CDNA5 / gfx1250, wave32. Pairwise arc-visibility radius kernel.
// Per 16x16 (i,j) tile: 3x V_WMMA_F32_16X16X4_F32 compute sq-dist, rx, ry as
// rank-4 bilinear forms; VALU does v_sqrt + one-compare cone-mask + fmac.
// Split-K over j (8 chunks) -> 4096 independent waves for MI455X occupancy;
// deterministic partial-sum buffers in d_ws, summed by a finalize kernel.
// All branches enclosing WMMA are wave-uniform scalars (readfirstlane) so
// EXEC stays all-ones, as required by the CDNA5 WMMA ISA.

typedef __attribute__((ext_vector_type(2))) float v2f;
typedef __attribute__((ext_vector_type(8))) float v8f;

#define TAN_HALF_FOV 0.70020753820970977f   // tan(35 deg)
#define SEG_POS   1024                      // positions per LDS segment (8 KB)
#define SEG_TILES (SEG_POS / 16)
#define WAVES_PER_BLOCK 8
#define THREADS (WAVES_PER_BLOCK * 32)

template<bool CHECK_DIAG>
__device__ __forceinline__ void process_tile(
    const float2* __restrict__ spos, int lt, int jt,
    int m, bool hi, int ibase,
    v2f A_sq, v2f A_rx, v2f A_ry,
    v8f& s_acc, v8f& c_acc)
{
    float2 pj = spos[(lt << 4) + m];        // ds_load_b64, 2-lane broadcast
    float  nj = pj.x * pj.x + pj.y * pj.y;

    // B 4x16 f32: low lanes supply rows K=0,1; high lanes rows K=2,3.
    v2f B_sq, B_xy;
    B_sq[0] = hi ? 1.0f : -2.0f * pj.x;
    B_sq[1] = hi ? nj   : -2.0f * pj.y;
    B_xy[0] = hi ? 1.0f : pj.x;
    B_xy[1] = hi ? 0.0f : pj.y;

    v8f zero = {};
    // sq[i,j] = |pi|^2 + |pj|^2 - 2 pi.pj
    v8f sq = __builtin_amdgcn_wmma_f32_16x16x4_f32(
        false, A_sq, false, B_sq, (short)0, zero, false, false);
    // rx[i,j] = (pj - pi) . h_i ; ry[i,j] = (pj - pi) x h_i
    v8f rx = __builtin_amdgcn_wmma_f32_16x16x4_f32(
        false, A_rx, false, B_xy, (short)0, zero, false, false);
    v8f ry = __builtin_amdgcn_wmma_f32_16x16x4_f32(
        false, A_ry, false, B_xy, (short)0, zero, false, false);

    const int jg = (jt << 4) + m;
    #pragma unroll
    for (int r = 0; r < 8; ++r) {
        // v_max_num flushes any NaN sq (cancellation guard) to 0 -> d is never NaN
        float d = __builtin_amdgcn_sqrtf(fmaxf(sq[r], 0.0f));   // v_sqrt_f32
        float t = rx[r] * (float)TAN_HALF_FOV;
        // Single-compare cone test: rx<=0 => t<=0 => |ry|<t false, so the
        // rx>0 check is implied. fabs folds into the v_cmp abs modifier.
        // (Boundary ry==+-t differs from the ref only on a measure-zero set.)
        bool inS = fabsf(ry[r]) < t;
        if (CHECK_DIAG) inS = inS & (jg != ibase + r);
        float mf = inS ? 1.0f : 0.0f;       // one v_cndmask
        c_acc[r] += mf;                     // v_add
        s_acc[r] = fmaf(d, mf, s_acc[r]);   // v_fmac (masked-off d*0 == 0)
    }
}

__device__ __forceinline__ float finalize_radius(float sv, float cv) {
    float mean = (cv > 0.0f) ? (sv / fmaxf(cv, 1.0f)) : 0.0f;
    mean = fminf(fmaxf(mean, 0.2f), 5.0f);              // clip [MIN_D, MAX_D]
    return 0.5f + (mean - 0.2f) * (3.5f / 4.8f);        // MIN_R + reg*(MAX_R-MIN_R)
}

__global__ __launch_bounds__(THREADS) void arc_pass1(
    const float* __restrict__ past, const float* __restrict__ pos,
    const int* __restrict__ indexes, const float* __restrict__ radii,
    float* __restrict__ out, float* __restrict__ partS, float* __restrict__ partC,
    int n, int chunks, int tpc)
{
    __shared__ float2 spos[SEG_POS];        // 8 KB -> many blocks per WGP

    const int tid  = threadIdx.x;
    const int lane = tid & 31;
    // Wave-uniform scalars in SGPRs: enclosing control flow stays scalar,
    // EXEC remains all-ones for every WMMA.
    const int wave = __builtin_amdgcn_readfirstlane(tid >> 5);

    const int ntiles    = n >> 4;
    const int rowBlocks = ntiles;
    const int rbGroups  = (rowBlocks + WAVES_PER_BLOCK - 1) / WAVES_PER_BLOCK;

    const int chunk = blockIdx.x / rbGroups;            // j-chunk (uniform per block)
    const int rbG   = blockIdx.x % rbGroups;
    const int rb    = rbG * WAVES_PER_BLOCK + wave;     // 16-row block (SGPR)
    const bool active = rb < rowBlocks;                 // scalar predicate

    const int t0 = chunk * tpc;
    const int t1 = min(ntiles, t0 + tpc);

    const int  m  = lane & 15;
    const bool hi = lane >= 16;                         // lanes 16-31: K=2,3 of A/B
    int ibase = 0;
    v2f A_sq = {}, A_rx = {}, A_ry = {};

    if (active) {
        // ---- Per-row (loop-invariant) A-matrix setup ----
        const int im  = (rb << 4) + m;
        float pix = pos[2 * im], piy = pos[2 * im + 1];
        float dxh = pix - past[2 * im];
        float dyh = piy - past[2 * im + 1];
        float l2  = dxh * dxh + dyh * dyh;
        float inv = (l2 > 0.0f) ? __builtin_amdgcn_rsqf(l2) : 0.0f;
        float ch  = (l2 > 0.0f) ? dxh * inv : 1.0f;     // cos(heading)
        float sh  = (l2 > 0.0f) ? dyh * inv : 0.0f;     // sin(heading)
        float ni  = pix * pix + piy * piy;
        float dc  = pix * ch + piy * sh;                // p_i . h
        float dr  = piy * ch - pix * sh;                // p_i x h
        // A 16x4 f32 layout: lane m -> (K0,K1), lane m+16 -> (K2,K3)
        A_sq[0] = hi ?  ni  : pix;  A_sq[1] = hi ? 1.0f : piy;
        A_rx[0] = hi ? -dc  : ch;   A_rx[1] = hi ? 0.0f : sh;
        A_ry[0] = hi ? -dr  : -sh;  A_ry[1] = hi ? 0.0f : ch;
        ibase   = (rb << 4) + (hi ? 8 : 0);
    }

    v8f s_acc = {};                 // masked distance partials, C/D layout
    v8f c_acc = {};                 // masked counts, C/D layout

    // ---- Stream j-range of this chunk through LDS in 8 KB segments ----
    const float2* pos2 = (const float2*)pos;
    for (int ts = t0; ts < t1; ts += SEG_TILES) {
        const int te   = min(t1, ts + SEG_TILES);
        const int cntP = (te - ts) << 4;
        __syncthreads();            // previous segment fully consumed
        for (int t = tid; t < cntP; t += THREADS) spos[t] = pos2[(ts << 4) + t];
        __syncthreads();
        if (active) {
            // Diagonal tile (jt == rb) hoisted: at most one tile per wave
            // needs the j != i guard.
            const int dEnd = min(te, rb);               // [ts, dEnd): before diag
            #pragma unroll 2
            for (int jt = ts; jt < dEnd; ++jt)
                process_tile<false>(spos, jt - ts, jt, m, hi, ibase,
                                    A_sq, A_rx, A_ry, s_acc, c_acc);
            if (rb >= ts && rb < te)
                process_tile<true >(spos, rb - ts, rb, m, hi, ibase,
                                    A_sq, A_rx, A_ry, s_acc, c_acc);
            const int aBeg = max(ts, rb + 1);           // [aBeg, te): after diag
            #pragma unroll 2
            for (int jt = aBeg; jt < te; ++jt)
                process_tile<false>(spos, jt - ts, jt, m, hi, ibase,
                                    A_sq, A_rx, A_ry, s_acc, c_acc);
        }
    }

    // ---- Reduce 16 columns (lanes) within each 16-lane half ----
    #pragma unroll
    for (int r = 0; r < 8; ++r) {
        float sv = s_acc[r], cv = c_acc[r];
        sv += __shfl_xor(sv, 1);  cv += __shfl_xor(cv, 1);
        sv += __shfl_xor(sv, 2);  cv += __shfl_xor(cv, 2);
        sv += __shfl_xor(sv, 4);  cv += __shfl_xor(cv, 4);
        sv += __shfl_xor(sv, 8);  cv += __shfl_xor(cv, 8);
        s_acc[r] = sv;  c_acc[r] = cv;
    }

    // ---- Lanes 0 and 16 own 8 consecutive rows each ----
    if (active && m == 0) {
        #pragma unroll
        for (int r = 0; r < 8; ++r) {
            const int i = ibase + r;
            if (chunks == 1) {
                out[i] = indexes[i] ? finalize_radius(s_acc[r], c_acc[r]) : radii[i];
            } else {
                partS[(size_t)chunk * n + i] = s_acc[r];
                partC[(size_t)chunk * n + i] = c_acc[r];
            }
        }
    }
}

__global__ __launch_bounds__(256) void arc_finalize(
    const int* __restrict__ indexes, const float* __restrict__ radii,
    const float* __restrict__ partS, const float* __restrict__ partC,
    float* __restrict__ out, int n, int chunks)
{
    const int i = blockIdx.x * blockDim.x + threadIdx.x;
    if (i >= n) return;
    float s = 0.0f, c = 0.0f;
    for (int k = 0; k < chunks; ++k) {      // deterministic fixed-order sum
        s += partS[(size_t)k * n + i];
        c += partC[(size_t)k * n + i];
    }
    out[i] = indexes[i] ? finalize_radius(s, c) : radii[i];
}

extern "C" void kernel_launch(void* const* d_in, const int* in_sizes, int n_in,
                              void* d_out, int out_size, void* d_ws, size_t ws_size,
                              hipStream_t stream) {
    const float* past = (const float*)d_in[0];   // past_ped_positions [N,2] f32
    const float* pos  = (const float*)d_in[1];   // ped_positions      [N,2] f32
    const int*   idx  = (const int*)d_in[2];     // indexes (bool -> int)
    const float* rad  = (const float*)d_in[3];   // all_radii [N] f32
    float* out = (float*)d_out;

    const int n = in_sizes[0] / 2;               // N = 8192 (multiple of 16)
    const int ntiles    = n >> 4;
    const int rowBlocks = ntiles;
    const int rbGroups  = (rowBlocks + WAVES_PER_BLOCK - 1) / WAVES_PER_BLOCK;

    int chunks = 8;                              // split-K over j for occupancy
    if (ws_size < (size_t)chunks * n * 2 * sizeof(float)) chunks = 1;
    const int tpc = (ntiles + chunks - 1) / chunks;

    float* partS = (float*)d_ws;
    float* partC = partS + (size_t)chunks * n;

    arc_pass1<<<rbGroups * chunks, THREADS, 0, stream>>>(
        past, pos, idx, rad, out, partS, partC, n, chunks, tpc);
    if (chunks > 1)
        arc_finalize<<<(n + 255) / 256, 256, 0, stream>>>(idx, rad, partS, partC, out, n, chunks);
}